// RecipeGAT_10634339025217
// MI455X (gfx1250) — compile-verified
//
#include <hip/hip_runtime.h>
#include <math.h>

typedef __attribute__((ext_vector_type(2))) float v2f;
typedef __attribute__((ext_vector_type(8))) float v8f;

static constexpr int NN  = 40000;
static constexpr int EE  = 400000;
static constexpr int EP  = EE + NN;   // edges + self loops
static constexpr int GG  = 256;

#define KC        32                  // K-chunk staged per async step
#define A_STRIDE  36                  // padded LDS row stride (floats) -> conflict-free

// ---------------------------------------------------------------------------
// WMMA fp32 GEMM:  C[n x M] = A[n x K] * B[K x M]   (row-major, K%32==0, M%16==0)
// Block = 256 threads (8 waves), covers 128 rows x 16 cols.
// A staged via global_load_async_to_lds_b128 (double-buffered, ASYNCcnt),
// B panel [K x 16] staged once. 8 x v_wmma_f32_16x16x4_f32 per chunk.
// ---------------------------------------------------------------------------
__global__ __launch_bounds__(256)
void wmma_gemm(const float* __restrict__ A, const float* __restrict__ B,
               float* __restrict__ C, int n, int K, int M)
{
    __shared__ float ldsB[256 * 16];              // up to K=256
    __shared__ float ldsA[2 * 128 * A_STRIDE];    // double-buffered A chunks

    const int tid     = threadIdx.x;
    const int colBase = blockIdx.y * 16;
    const int row0blk = blockIdx.x * 128;

    // Issue one 128 x KC A-chunk as async b128 copies (exactly 4 instr / wave).
    auto issueA = [&](int k0, int buf) {
        for (int i = tid; i < 128 * (KC / 4); i += 256) {
            int row  = i >> 3;                    // 0..127
            int q    = i & 7;                     // 0..7 (16B units)
            int grow = row0blk + row;
            grow = (grow < n) ? grow : (n - 1);   // clamp: no divergence, no OOB
            const float* g = A + (size_t)grow * K + k0 + q * 4;
            const float* l = &ldsA[buf * (128 * A_STRIDE) + row * A_STRIDE + q * 4];
            unsigned loff = (unsigned)(size_t)l;  // low 32 bits = LDS byte offset
            asm volatile("global_load_async_to_lds_b128 %0, %1, off"
                         :: "v"(loff), "v"(g) : "memory");
        }
    };

    issueA(0, 0);

    // Stage B panel [K x 16] (regular loads; covered by first barrier).
    for (int i = tid; i < K * 16; i += 256) {
        int k = i >> 4, c = i & 15;
        ldsB[i] = B[(size_t)k * M + colBase + c];
    }

    const int wave = tid >> 5;
    const int lane = tid & 31;
    const int half = lane >> 4;                   // 0: K pair {0,1}, 1: K pair {2,3}
    const int l16  = lane & 15;
    const int lrow = wave * 16 + l16;             // block-local A row for this lane

    v8f acc = {};
    const int nch = K / KC;
    for (int c = 0; c < nch; ++c) {
        if (c + 1 < nch) {
            issueA((c + 1) * KC, (c + 1) & 1);
            asm volatile("s_wait_asynccnt 0x4" ::: "memory");   // chunk c landed
        } else {
            asm volatile("s_wait_asynccnt 0x0" ::: "memory");
        }
        __syncthreads();

        const float* Ab = &ldsA[(c & 1) * (128 * A_STRIDE) + lrow * A_STRIDE];
        const int kB = c * KC;
#pragma unroll
        for (int kk = 0; kk < KC; kk += 4) {
            v2f a, b;
            a.x = Ab[kk + 2 * half];                          // A 16x4 frag
            a.y = Ab[kk + 2 * half + 1];
            b.x = ldsB[(kB + kk + 2 * half) * 16 + l16];      // B 4x16 frag
            b.y = ldsB[(kB + kk + 2 * half + 1) * 16 + l16];
            acc = __builtin_amdgcn_wmma_f32_16x16x4_f32(
                      false, a, false, b, (short)0, acc, false, false);
        }
        __syncthreads();   // protect double buffer before chunk c+2 is issued
    }

    // C/D layout: VGPR i -> (M = i + 8*half, N = lane&15)
#pragma unroll
    for (int i = 0; i < 8; ++i) {
        int r = row0blk + wave * 16 + half * 8 + i;
        if (r < n) C[(size_t)r * M + colBase + l16] = acc[i];
    }
}

// ---------------------------------------------------------------------------
// Per-(node, head) attention score dots: s = <h[n,h,:], a[h,:]>
// ---------------------------------------------------------------------------
template <int HEADS>
__global__ void scores_kernel(const float* __restrict__ h,
                              const float* __restrict__ a_src,
                              const float* __restrict__ a_dst,
                              float* __restrict__ s_src,
                              float* __restrict__ s_dst, int n)
{
    int idx = blockIdx.x * blockDim.x + threadIdx.x;
    if (idx >= n * HEADS) return;
    int node = idx / HEADS, hd = idx % HEADS;
    const float* row = h + (size_t)node * (HEADS * 128) + hd * 128;
    const float* as  = a_src + hd * 128;
    const float* ad  = a_dst + hd * 128;
    float ss = 0.f, sd = 0.f;
    for (int c = 0; c < 128; c += 4) {
        float4 v = *reinterpret_cast<const float4*>(row + c);
        float4 s = *reinterpret_cast<const float4*>(as + c);
        float4 d = *reinterpret_cast<const float4*>(ad + c);
        ss += v.x * s.x + v.y * s.y + v.z * s.z + v.w * s.w;
        sd += v.x * d.x + v.y * d.y + v.z * d.z + v.w * d.w;
    }
    s_src[idx] = ss;
    s_dst[idx] = sd;
}

// ---------------------------------------------------------------------------
// CSR build over dst (shared by all 3 layers; edges E..E+N-1 are self loops)
// ---------------------------------------------------------------------------
__global__ void count_deg(const int* __restrict__ dst, int* __restrict__ deg,
                          int e, int n)
{
    int i = blockIdx.x * blockDim.x + threadIdx.x;
    if (i >= e + n) return;
    int d = (i < e) ? dst[i] : (i - e);
    atomicAdd(&deg[d], 1);
}

__global__ void scan_partial(const int* __restrict__ deg, int* __restrict__ excl,
                             int* __restrict__ bsum, int n)
{
    __shared__ int sh[256];
    int i = blockIdx.x * 256 + threadIdx.x;
    int v = (i < n) ? deg[i] : 0;
    sh[threadIdx.x] = v;
    __syncthreads();
    for (int off = 1; off < 256; off <<= 1) {
        int t = (threadIdx.x >= off) ? sh[threadIdx.x - off] : 0;
        __syncthreads();
        sh[threadIdx.x] += t;
        __syncthreads();
    }
    if (i < n) excl[i] = sh[threadIdx.x] - v;                 // exclusive
    if (threadIdx.x == 255) bsum[blockIdx.x] = sh[255];
}

__global__ void scan_sums(int* bsum, int nb)   // single block, nb <= 256
{
    __shared__ int sh[256];
    int v = (threadIdx.x < nb) ? bsum[threadIdx.x] : 0;
    sh[threadIdx.x] = v;
    __syncthreads();
    for (int off = 1; off < 256; off <<= 1) {
        int t = (threadIdx.x >= off) ? sh[threadIdx.x - off] : 0;
        __syncthreads();
        sh[threadIdx.x] += t;
        __syncthreads();
    }
    if (threadIdx.x < nb) bsum[threadIdx.x] = sh[threadIdx.x] - v;  // exclusive
}

__global__ void scan_finalize(const int* __restrict__ excl, const int* __restrict__ bsum,
                              int* __restrict__ row_ptr, int* __restrict__ cursor,
                              int n, int total)
{
    int i = blockIdx.x * blockDim.x + threadIdx.x;
    if (i < n) {
        int v = excl[i] + bsum[i >> 8];
        row_ptr[i] = v;
        cursor[i]  = v;
    }
    if (i == 0) row_ptr[n] = total;
}

__global__ void scatter_edges(const int* __restrict__ src, const int* __restrict__ dst,
                              int* __restrict__ cursor, int* __restrict__ src_sorted,
                              int* __restrict__ eid_sorted, int e, int n)
{
    int i = blockIdx.x * blockDim.x + threadIdx.x;
    if (i >= e + n) return;
    int s, d;
    if (i < e) { s = src[i]; d = dst[i]; }
    else       { s = d = i - e; }
    int pos = atomicAdd(&cursor[d], 1);
    src_sorted[pos] = s;
    eid_sorted[pos] = i;
}

// ---------------------------------------------------------------------------
// Segment softmax: one wave32 per dst node (shuffle reductions, no atomics)
// ---------------------------------------------------------------------------
template <int HEADS>
__global__ __launch_bounds__(256)
void edge_softmax(const int* __restrict__ row_ptr, const int* __restrict__ src_sorted,
                  const int* __restrict__ eid_sorted,
                  const float* __restrict__ s_src, const float* __restrict__ s_dst,
                  float* __restrict__ alpha, int n)
{
    int node = blockIdx.x * 8 + (threadIdx.x >> 5);
    if (node >= n) return;
    int lane  = threadIdx.x & 31;
    int start = row_ptr[node], end = row_ptr[node + 1];

    float sd[HEADS], mx[HEADS], zs[HEADS];
#pragma unroll
    for (int h = 0; h < HEADS; ++h) { sd[h] = s_dst[node * HEADS + h]; mx[h] = -1e30f; zs[h] = 0.f; }

    for (int i = start + lane; i < end; i += 32) {
        int s = src_sorted[i];
#pragma unroll
        for (int h = 0; h < HEADS; ++h) {
            float el = s_src[s * HEADS + h] + sd[h];
            el = (el > 0.f) ? el : 0.2f * el;          // leaky_relu
            mx[h] = fmaxf(mx[h], el);
        }
    }
#pragma unroll
    for (int h = 0; h < HEADS; ++h)
        for (int off = 16; off; off >>= 1) mx[h] = fmaxf(mx[h], __shfl_xor(mx[h], off, 32));

    for (int i = start + lane; i < end; i += 32) {
        int s = src_sorted[i];
        int e = eid_sorted[i];
#pragma unroll
        for (int h = 0; h < HEADS; ++h) {
            float el = s_src[s * HEADS + h] + sd[h];
            el = (el > 0.f) ? el : 0.2f * el;
            float p = __expf(el - mx[h]);
            alpha[(size_t)e * HEADS + h] = p;
            zs[h] += p;
        }
    }
#pragma unroll
    for (int h = 0; h < HEADS; ++h)
        for (int off = 16; off; off >>= 1) zs[h] += __shfl_xor(zs[h], off, 32);

    float inv[HEADS];
#pragma unroll
    for (int h = 0; h < HEADS; ++h) inv[h] = 1.f / (zs[h] + 1e-16f);

    for (int i = start + lane; i < end; i += 32) {
        int e = eid_sorted[i];
#pragma unroll
        for (int h = 0; h < HEADS; ++h) alpha[(size_t)e * HEADS + h] *= inv[h];
    }
}

// ---------------------------------------------------------------------------
// Aggregation (H=4): block per node, gather over incoming edges, fused
// mean-over-heads + bias + ELU + BatchNorm epilogue. Reads hit L2 (82MB < 192MB).
// ---------------------------------------------------------------------------
__global__ __launch_bounds__(256)
void aggregate4(const int* __restrict__ row_ptr, const int* __restrict__ src_sorted,
                const int* __restrict__ eid_sorted, const float* __restrict__ alpha,
                const float* __restrict__ hfeat, const float* __restrict__ bias,
                const float* __restrict__ bn_g, const float* __restrict__ bn_b,
                const float* __restrict__ bn_m, const float* __restrict__ bn_v,
                float* __restrict__ out, int n)
{
    __shared__ float sh[512];
    int node = blockIdx.x;
    int tid  = threadIdx.x;
    int f0 = tid, f1 = tid + 256;
    int h0 = f0 >> 7, h1 = f1 >> 7;
    int start = row_ptr[node], end = row_ptr[node + 1];
    float acc0 = 0.f, acc1 = 0.f;
    for (int i = start; i < end; ++i) {
        int s = src_sorted[i];
        int e = eid_sorted[i];
        if (i + 1 < end)
            __builtin_prefetch(hfeat + (size_t)src_sorted[i + 1] * 512 + f0, 0, 0);
        const float* hr = hfeat + (size_t)s * 512;
        acc0 += alpha[(size_t)e * 4 + h0] * hr[f0];
        acc1 += alpha[(size_t)e * 4 + h1] * hr[f1];
    }
    sh[f0] = acc0; sh[f1] = acc1;
    __syncthreads();
    if (tid < 128) {
        float m = 0.25f * (sh[tid] + sh[tid + 128] + sh[tid + 256] + sh[tid + 384]) + bias[tid];
        m = (m > 0.f) ? m : (__expf(m) - 1.f);                               // ELU
        m = (m - bn_m[tid]) * rsqrtf(bn_v[tid] + 1e-5f) * bn_g[tid] + bn_b[tid];  // BN (eval)
        out[(size_t)node * 128 + tid] = m;
    }
}

// Aggregation (H=1, layer 3): no ELU/BN, bias only.
__global__ __launch_bounds__(128)
void aggregate1(const int* __restrict__ row_ptr, const int* __restrict__ src_sorted,
                const int* __restrict__ eid_sorted, const float* __restrict__ alpha,
                const float* __restrict__ hfeat, const float* __restrict__ bias,
                float* __restrict__ out, int n)
{
    int node = blockIdx.x;
    int c    = threadIdx.x;           // 0..127
    int start = row_ptr[node], end = row_ptr[node + 1];
    float acc = 0.f;
    for (int i = start; i < end; ++i) {
        int s = src_sorted[i];
        int e = eid_sorted[i];
        acc += alpha[e] * hfeat[(size_t)s * 128 + c];
    }
    out[(size_t)node * 128 + c] = acc + bias[c];
}

// ---------------------------------------------------------------------------
// Global mean pool
// ---------------------------------------------------------------------------
__global__ void pool_scatter(const float* __restrict__ h, const int* __restrict__ batch,
                             float* __restrict__ tot, int n)
{
    int i = blockIdx.x * blockDim.x + threadIdx.x;
    if (i >= n * 128) return;
    int node = i >> 7, c = i & 127;
    atomicAdd(&tot[batch[node] * 128 + c], h[i]);
}

__global__ void pool_count(const int* __restrict__ batch, float* __restrict__ cnt, int n)
{
    int i = blockIdx.x * blockDim.x + threadIdx.x;
    if (i < n) atomicAdd(&cnt[batch[i]], 1.0f);
}

__global__ void pool_final(const float* __restrict__ tot, const float* __restrict__ cnt,
                           float* __restrict__ out)
{
    int i = blockIdx.x * blockDim.x + threadIdx.x;   // G*128
    if (i >= GG * 128) return;
    out[i] = tot[i] / fmaxf(cnt[i >> 7], 1.0f);
}

// ---------------------------------------------------------------------------
extern "C" void kernel_launch(void* const* d_in, const int* in_sizes, int n_in,
                              void* d_out, int out_size, void* d_ws, size_t ws_size,
                              hipStream_t stream)
{
    (void)in_sizes; (void)n_in; (void)out_size; (void)ws_size;

    const float* x   = (const float*)d_in[0];
    const int*   src = (const int*)d_in[1];
    const int*   dst = (const int*)d_in[2];
    const int*   bat = (const int*)d_in[3];
    const float* W1  = (const float*)d_in[4];
    const float* as1 = (const float*)d_in[5];
    const float* ad1 = (const float*)d_in[6];
    const float* b1  = (const float*)d_in[7];
    const float* g1  = (const float*)d_in[8];
    const float* be1 = (const float*)d_in[9];
    const float* m1  = (const float*)d_in[10];
    const float* v1  = (const float*)d_in[11];
    const float* W2  = (const float*)d_in[12];
    const float* as2 = (const float*)d_in[13];
    const float* ad2 = (const float*)d_in[14];
    const float* b2  = (const float*)d_in[15];
    const float* g2  = (const float*)d_in[16];
    const float* be2 = (const float*)d_in[17];
    const float* m2  = (const float*)d_in[18];
    const float* v2  = (const float*)d_in[19];
    const float* W3  = (const float*)d_in[20];
    const float* as3 = (const float*)d_in[21];
    const float* ad3 = (const float*)d_in[22];
    const float* b3  = (const float*)d_in[23];

    char* p = (char*)d_ws;
    auto carve = [&](size_t bytes) {
        void* r = (void*)p;
        p += (bytes + 255) & ~size_t(255);
        return r;
    };
    float* hbig   = (float*)carve((size_t)NN * 512 * sizeof(float)); // h1/h2 features, later final node feats
    float* hb1    = (float*)carve((size_t)NN * 128 * sizeof(float)); // post-layer1, later layer3 features
    float* hb2    = (float*)carve((size_t)NN * 128 * sizeof(float)); // post-layer2
    float* ssrc   = (float*)carve((size_t)NN * 4 * sizeof(float));
    float* sdst   = (float*)carve((size_t)NN * 4 * sizeof(float));
    float* alpha  = (float*)carve((size_t)EP * 4 * sizeof(float));
    int*   deg    = (int*)carve((size_t)NN * sizeof(int));
    int*   excl   = (int*)carve((size_t)NN * sizeof(int));
    int*   bsum   = (int*)carve(1024);
    int*   rowp   = (int*)carve((size_t)(NN + 1) * sizeof(int));
    int*   cursor = (int*)carve((size_t)NN * sizeof(int));
    int*   ssort  = (int*)carve((size_t)EP * sizeof(int));
    int*   esort  = (int*)carve((size_t)EP * sizeof(int));
    float* pool   = (float*)carve((size_t)GG * 128 * sizeof(float));
    float* cnt    = (float*)carve((size_t)GG * sizeof(float));

    const int tpb = 256;
    const int nb  = (NN + 255) / 256;

    // ---- CSR by dst (shared by all layers) ----
    hipMemsetAsync(deg, 0, NN * sizeof(int), stream);
    count_deg<<<(EP + tpb - 1) / tpb, tpb, 0, stream>>>(dst, deg, EE, NN);
    scan_partial<<<nb, 256, 0, stream>>>(deg, excl, bsum, NN);
    scan_sums<<<1, 256, 0, stream>>>(bsum, nb);
    scan_finalize<<<nb, 256, 0, stream>>>(excl, bsum, rowp, cursor, NN, EP);
    scatter_edges<<<(EP + tpb - 1) / tpb, tpb, 0, stream>>>(src, dst, cursor, ssort, esort, EE, NN);

    // ---- Layer 1: GAT(256 -> 4x128) + ELU + BN ----
    {
        dim3 grid((NN + 127) / 128, 512 / 16);
        wmma_gemm<<<grid, 256, 0, stream>>>(x, W1, hbig, NN, 256, 512);
    }
    scores_kernel<4><<<(NN * 4 + 255) / 256, 256, 0, stream>>>(hbig, as1, ad1, ssrc, sdst, NN);
    edge_softmax<4><<<(NN + 7) / 8, 256, 0, stream>>>(rowp, ssort, esort, ssrc, sdst, alpha, NN);
    aggregate4<<<NN, 256, 0, stream>>>(rowp, ssort, esort, alpha, hbig, b1, g1, be1, m1, v1, hb1, NN);

    // ---- Layer 2: GAT(128 -> 4x128) + ELU + BN ----
    {
        dim3 grid((NN + 127) / 128, 512 / 16);
        wmma_gemm<<<grid, 256, 0, stream>>>(hb1, W2, hbig, NN, 128, 512);
    }
    scores_kernel<4><<<(NN * 4 + 255) / 256, 256, 0, stream>>>(hbig, as2, ad2, ssrc, sdst, NN);
    edge_softmax<4><<<(NN + 7) / 8, 256, 0, stream>>>(rowp, ssort, esort, ssrc, sdst, alpha, NN);
    aggregate4<<<NN, 256, 0, stream>>>(rowp, ssort, esort, alpha, hbig, b2, g2, be2, m2, v2, hb2, NN);

    // ---- Layer 3: GAT(128 -> 1x128), bias only ----
    {
        dim3 grid((NN + 127) / 128, 128 / 16);
        wmma_gemm<<<grid, 256, 0, stream>>>(hb2, W3, hb1, NN, 128, 128);   // hb1 := h3 features
    }
    scores_kernel<1><<<(NN + 255) / 256, 256, 0, stream>>>(hb1, as3, ad3, ssrc, sdst, NN);
    edge_softmax<1><<<(NN + 7) / 8, 256, 0, stream>>>(rowp, ssort, esort, ssrc, sdst, alpha, NN);
    aggregate1<<<NN, 128, 0, stream>>>(rowp, ssort, esort, alpha, hb1, b3, hbig, NN);  // hbig := node outputs

    // ---- Global mean pool ----
    hipMemsetAsync(pool, 0, GG * 128 * sizeof(float), stream);
    hipMemsetAsync(cnt, 0, GG * sizeof(float), stream);
    pool_scatter<<<(NN * 128 + 255) / 256, 256, 0, stream>>>(hbig, bat, pool, NN);
    pool_count<<<(NN + 255) / 256, 256, 0, stream>>>(bat, cnt, NN);
    pool_final<<<(GG * 128 + 255) / 256, 256, 0, stream>>>(pool, cnt, (float*)d_out);
}